// ReviewRepresentationConv_33672543601279
// MI455X (gfx1250) — compile-verified
//
#include <hip/hip_runtime.h>
#include <math.h>

#define N_NODES_C 50000
#define N_EDGES_C 800000
#define DIM 128
#define NEG_SLOPE 0.01f

typedef __attribute__((ext_vector_type(2))) float v2f;
typedef __attribute__((ext_vector_type(8))) float v8f;

// ---------------------------------------------------------------- zero fill
__global__ void zero_kernel(float* __restrict__ p, long n) {
    long i = (long)blockIdx.x * blockDim.x + threadIdx.x;
    long stride = (long)gridDim.x * blockDim.x;
    for (; i < n; i += stride) p[i] = 0.0f;
}

// ------------------------------------------- scatter mean: h_sum[dst]+=x[src]
// one wave32 per edge; lane handles 4 contiguous dims (float4 = b128 load)
__global__ void scatter_mean_kernel(const float* __restrict__ x,
                                    const int* __restrict__ src_idx,
                                    const int* __restrict__ dst_idx,
                                    float* __restrict__ hsum,
                                    float* __restrict__ deg,
                                    int nEdges) {
    int e = blockIdx.x * blockDim.y + threadIdx.y;
    if (e >= nEdges) return;                 // wave-uniform exit
    int lane = threadIdx.x;                  // 0..31
    int s = src_idx[e];
    int d = dst_idx[e];
    const float4 v = *(const float4*)(x + (size_t)s * DIM + lane * 4);
    float* hp = hsum + (size_t)d * DIM + lane * 4;
    atomicAdd(hp + 0, v.x);
    atomicAdd(hp + 1, v.y);
    atomicAdd(hp + 2, v.z);
    atomicAdd(hp + 3, v.w);
    if (lane == 0) atomicAdd(deg + d, 1.0f);
}

// ------------------------------------------------------- h_mean = h_sum/deg
__global__ void mean_div_kernel(float* __restrict__ hsum,
                                const float* __restrict__ deg,
                                int nNodes) {
    long total = (long)nNodes * DIM;
    long i = (long)blockIdx.x * blockDim.x + threadIdx.x;
    long stride = (long)gridDim.x * blockDim.x;
    for (; i < total; i += stride) {
        float dg = deg[i >> 7];              // DIM == 128
        hsum[i] = hsum[i] / fmaxf(dg, 1.0f);
    }
}

// ------------------------------------------------- node GEMM via f32 WMMA
// out[M,128] = A[M,128] @ W[128,128] + bias, one 16x16 tile per wave,
// K=128 as 32 chained V_WMMA_F32_16X16X4_F32 ops.
__global__ void gemm_wmma_f32(const float* __restrict__ A,
                              const float* __restrict__ W,
                              const float* __restrict__ bias,
                              float* __restrict__ out,
                              int M) {
    const int mtiles = (M + 15) >> 4;
    const int mtile = blockIdx.x * blockDim.y + threadIdx.y;
    if (mtile >= mtiles) return;             // wave-uniform: EXEC stays all-1s
    const int nbase = blockIdx.y * 16;
    const int lane = threadIdx.x;            // 0..31
    const int half = lane >> 4;              // 0: K={0,1}, 1: K={2,3}
    const int l15 = lane & 15;

    const int mrow = mtile * 16 + l15;
    const int mclamp = (mrow < M) ? mrow : (M - 1);   // safe tail loads
    const float* __restrict__ arow = A + (size_t)mclamp * DIM;
    const int ncol = nbase + l15;

    v8f c = {};
#pragma unroll
    for (int kk = 0; kk < DIM; kk += 4) {
        const int ka = kk + half * 2;
        v2f a, b;
        a[0] = arow[ka];                     // A[m][ka], A[m][ka+1] -> b64 load
        a[1] = arow[ka + 1];
        b[0] = W[(size_t)ka * DIM + ncol];   // B[k][n] column fragment
        b[1] = W[(size_t)(ka + 1) * DIM + ncol];
        // 8 args: (neg_a, A, neg_b, B, c_mod, C, reuse_a, reuse_b)
        c = __builtin_amdgcn_wmma_f32_16x16x4_f32(
                false, a, false, b, (short)0, c, false, false);
    }

    const float bv = bias[ncol];
#pragma unroll
    for (int r = 0; r < 8; ++r) {
        // C/D layout: VGPR r -> row (r + half*8), col = l15
        int row = mtile * 16 + r + half * 8;
        if (row < M) out[(size_t)row * DIM + ncol] = c[r] + bv;
    }
}

// ---------------- per-edge score: s = exp(dot(lrelu(haS[src]+haD[dst]), w)+b)
__global__ void edge_score_kernel(const float* __restrict__ ha_src,
                                  const float* __restrict__ ha_dst,
                                  const int* __restrict__ src_idx,
                                  const int* __restrict__ dst_idx,
                                  const float* __restrict__ att_w,
                                  const float* __restrict__ att_b,
                                  float* __restrict__ s_out,
                                  float* __restrict__ denom,
                                  int nEdges) {
    int e = blockIdx.x * blockDim.y + threadIdx.y;
    if (e >= nEdges) return;
    int lane = threadIdx.x;
    int s = src_idx[e];
    int d = dst_idx[e];
    float4 vs = *(const float4*)(ha_src + (size_t)s * DIM + lane * 4);
    float4 vd = *(const float4*)(ha_dst + (size_t)d * DIM + lane * 4);
    float4 w  = *(const float4*)(att_w + lane * 4);
    float p = 0.0f, t;
    t = vs.x + vd.x; p += (t > 0.0f ? t : t * NEG_SLOPE) * w.x;
    t = vs.y + vd.y; p += (t > 0.0f ? t : t * NEG_SLOPE) * w.y;
    t = vs.z + vd.z; p += (t > 0.0f ? t : t * NEG_SLOPE) * w.z;
    t = vs.w + vd.w; p += (t > 0.0f ? t : t * NEG_SLOPE) * w.w;
    // wave32 tree reduction
#pragma unroll
    for (int m = 16; m > 0; m >>= 1) p += __shfl_xor(p, m, 32);
    if (lane == 0) {
        float val = expf(p + att_b[0]);
        s_out[e] = val;
        atomicAdd(denom + d, val);
    }
}

// ------------------------- out[dst] += x[src] * (s[e] / denom[dst])
__global__ void edge_aggregate_kernel(const float* __restrict__ x,
                                      const int* __restrict__ src_idx,
                                      const int* __restrict__ dst_idx,
                                      const float* __restrict__ s_in,
                                      const float* __restrict__ denom,
                                      float* __restrict__ out,
                                      int nEdges) {
    int e = blockIdx.x * blockDim.y + threadIdx.y;
    if (e >= nEdges) return;
    int lane = threadIdx.x;
    int s = src_idx[e];
    int d = dst_idx[e];
    float a = s_in[e] / denom[d];
    float4 v = *(const float4*)(x + (size_t)s * DIM + lane * 4);
    float* op = out + (size_t)d * DIM + lane * 4;
    atomicAdd(op + 0, v.x * a);
    atomicAdd(op + 1, v.y * a);
    atomicAdd(op + 2, v.z * a);
    atomicAdd(op + 3, v.w * a);
}

// ---------------------------------------------------------------------------
extern "C" void kernel_launch(void* const* d_in, const int* in_sizes, int n_in,
                              void* d_out, int out_size, void* d_ws, size_t ws_size,
                              hipStream_t stream) {
    const float* x     = (const float*)d_in[0];
    const int*   src   = (const int*)  d_in[1];
    const int*   dst   = (const int*)  d_in[2];
    const float* src_w = (const float*)d_in[3];
    const float* src_b = (const float*)d_in[4];
    const float* dst_w = (const float*)d_in[5];
    const float* dst_b = (const float*)d_in[6];
    const float* att_w = (const float*)d_in[7];
    const float* att_b = (const float*)d_in[8];
    float* out = (float*)d_out;

    const int nNodes = N_NODES_C;
    const int nEdges = N_EDGES_C;

    // workspace layout (floats): deg | h_sum/h_mean | denom | ha_src | ha_dst | s
    float* ws = (float*)d_ws;
    size_t off = 0;
    float* deg   = ws + off; off += 50048;                    // padded 50000
    float* hbuf  = ws + off; off += (size_t)nNodes * DIM;
    float* denom = ws + off; off += 50048;
    float* haS   = ws + off; off += (size_t)nNodes * DIM;
    float* haD   = ws + off; off += (size_t)nNodes * DIM;
    float* sArr  = ws + off; off += (size_t)nEdges;

    // zero atomic accumulators (deg, hbuf, denom are contiguous) and output
    long zcount = 50048 + (long)nNodes * DIM + 50048;
    zero_kernel<<<1024, 256, 0, stream>>>(ws, zcount);
    zero_kernel<<<1024, 256, 0, stream>>>(out, (long)nNodes * DIM);

    // edge-parallel launches: wave32 per edge, 8 waves per block
    dim3 eb(32, 8);
    int eg = (nEdges + 7) / 8;

    scatter_mean_kernel<<<eg, eb, 0, stream>>>(x, src, dst, hbuf, deg, nEdges);
    mean_div_kernel<<<1024, 256, 0, stream>>>(hbuf, deg, nNodes);

    // node GEMMs with f32 WMMA: grid = (M tiles / 4 waves, 8 N tiles)
    {
        int mtiles = (nNodes + 15) / 16;
        dim3 gb(32, 4);
        dim3 gg((mtiles + 3) / 4, DIM / 16);
        gemm_wmma_f32<<<gg, gb, 0, stream>>>(x,    src_w, src_b, haS, nNodes);
        gemm_wmma_f32<<<gg, gb, 0, stream>>>(hbuf, dst_w, dst_b, haD, nNodes);
    }

    edge_score_kernel<<<eg, eb, 0, stream>>>(haS, haD, src, dst, att_w, att_b,
                                             sArr, denom, nEdges);
    edge_aggregate_kernel<<<eg, eb, 0, stream>>>(x, src, dst, sArr, denom,
                                                 out, nEdges);
}